// Model_25185688224518
// MI455X (gfx1250) — compile-verified
//
#include <hip/hip_runtime.h>

// ---------------------------------------------------------------------------
// GIN encoder (3 layers) for gfx1250: scatter-add aggregation (HBM bound),
// WMMA f32 16x16x4 GEMMs, fused BN stats + normalize + graph pooling.
// ---------------------------------------------------------------------------

#define N_NODES  50000
#define N_EDGES  600000
#define DIM      128
#define N_LAYERS 3
#define N_GRAPHS 512
#define BN_EPS   1e-5f

typedef __attribute__((ext_vector_type(2))) float v2f;
typedef __attribute__((ext_vector_type(8))) float v8f;

// ---------------------------------------------------------------------------
// zero a float buffer (grid-stride)
// ---------------------------------------------------------------------------
__global__ __launch_bounds__(256) void gin_zero(float* __restrict__ p, int n) {
    int i = blockIdx.x * blockDim.x + threadIdx.x;
    int stride = gridDim.x * blockDim.x;
    for (; i < n; i += stride) p[i] = 0.0f;
}

// ---------------------------------------------------------------------------
// agg = h (self term of GIN, eps=0)
// ---------------------------------------------------------------------------
__global__ __launch_bounds__(256) void gin_copy(const float* __restrict__ src,
                                                float* __restrict__ dst, int n4) {
    int i = blockIdx.x * blockDim.x + threadIdx.x;
    if (i < n4) {
        float4 v = reinterpret_cast<const float4*>(src)[i];
        reinterpret_cast<float4*>(dst)[i] = v;
    }
}

// ---------------------------------------------------------------------------
// agg[dst] += h[src] : one wave (32 lanes) per edge, float4 per lane (128 f32)
// ---------------------------------------------------------------------------
__global__ __launch_bounds__(256) void gin_scatter(const float* __restrict__ h,
                                                   const int* __restrict__ src,
                                                   const int* __restrict__ dst,
                                                   float* __restrict__ agg,
                                                   int n_edges) {
    int gid  = blockIdx.x * blockDim.x + threadIdx.x;
    int e    = gid >> 5;
    int lane = gid & 31;
    if (e >= n_edges) return;
    int s = src[e];
    int d = dst[e];
    const float4 v = *reinterpret_cast<const float4*>(h + (size_t)s * DIM + lane * 4);
    float* out = agg + (size_t)d * DIM + lane * 4;
    atomicAdd(out + 0, v.x);
    atomicAdd(out + 1, v.y);
    atomicAdd(out + 2, v.z);
    atomicAdd(out + 3, v.w);
}

// ---------------------------------------------------------------------------
// Out = relu(A @ W + bias), optionally accumulate per-column sum / sumsq.
// A: [M x 128], W: [128 x 128] row-major, M = 3125 * 16 exactly.
// One wave computes a 16x16 tile; wave w of the block handles N-tile w (0..7);
// block handles one 16-row strip. K marched in steps of 4 via
// V_WMMA_F32_16X16X4_F32. EXEC is all-1s throughout (no divergence).
//
// Fragment layouts (CDNA5 ISA 7.12.2, 32-bit operands, wave32):
//   A 16x4:  lane -> m = lane%16, khalf = lane/16; vgpr j -> K = 2*khalf + j
//   B 4x16:  lane -> n = lane%16, khalf = lane/16; vgpr j -> K = 2*khalf + j
//   C 16x16: vgpr r -> M = r + 8*(lane/16), N = lane%16
// ---------------------------------------------------------------------------
__global__ __launch_bounds__(256) void gin_gemm_relu(const float* __restrict__ A,
                                                     const float* __restrict__ W,
                                                     const float* __restrict__ bias,
                                                     float* __restrict__ Out,
                                                     float* __restrict__ colsum,
                                                     float* __restrict__ colsumsq) {
    const int lane  = threadIdx.x & 31;
    const int wave  = threadIdx.x >> 5;     // 0..7 -> N tile
    const int mtile = blockIdx.x;           // 0..3124
    const int m     = lane & 15;
    const int khalf = lane >> 4;            // 0 or 1
    const int ncol  = wave * 16 + m;        // global column this lane owns in B/C

    const float* Arow = A + (size_t)(mtile * 16 + m) * DIM;

    // accumulator initialized with bias: all 8 C values of a lane share ncol
    const float bv = bias[ncol];
    v8f c;
#pragma unroll
    for (int r = 0; r < 8; ++r) c[r] = bv;

#pragma unroll 4
    for (int kk = 0; kk < DIM; kk += 4) {
        const int k = kk + 2 * khalf;
        v2f a = *reinterpret_cast<const v2f*>(Arow + k);   // A[m][k], A[m][k+1]
        v2f b;
        b.x = W[(size_t)k * DIM + ncol];                   // W[k][ncol]
        b.y = W[(size_t)(k + 1) * DIM + ncol];             // W[k+1][ncol]
        c = __builtin_amdgcn_wmma_f32_16x16x4_f32(
                /*neg_a=*/false, a, /*neg_b=*/false, b,
                /*c_mod=*/(short)0, c, /*reuse_a=*/false, /*reuse_b=*/false);
    }

    // ReLU + store + (optional) per-column BN statistics
    float s = 0.0f, s2 = 0.0f;
#pragma unroll
    for (int r = 0; r < 8; ++r) {
        float v = c[r] > 0.0f ? c[r] : 0.0f;
        const int row = mtile * 16 + r + 8 * khalf;
        Out[(size_t)row * DIM + ncol] = v;
        s  += v;
        s2 += v * v;
    }
    if (colsum != nullptr) {
        atomicAdd(&colsum[ncol], s);
        atomicAdd(&colsumsq[ncol], s2);
    }
}

// ---------------------------------------------------------------------------
// scale/shift from accumulated stats (biased variance, training-mode BN)
// ---------------------------------------------------------------------------
__global__ __launch_bounds__(128) void gin_bnparams(const float* __restrict__ stats,
                                                    const float* __restrict__ gamma,
                                                    const float* __restrict__ beta,
                                                    float* __restrict__ scaleshift) {
    const int j = threadIdx.x;                 // 0..127
    const float inv_n = 1.0f / (float)N_NODES;
    const float mean  = stats[j] * inv_n;
    const float var   = stats[DIM + j] * inv_n - mean * mean;
    const float sc    = gamma[j] * rsqrtf(var + BN_EPS);
    scaleshift[j]       = sc;
    scaleshift[DIM + j] = beta[j] - mean * sc;
}

// ---------------------------------------------------------------------------
// h = h2*scale + shift ; pool[batch[node]] += h   (pool row stride = 3*DIM)
// one wave per node, float4 per lane
// ---------------------------------------------------------------------------
__global__ __launch_bounds__(256) void gin_bn_pool(const float* __restrict__ h2,
                                                   const float* __restrict__ ss,
                                                   const int* __restrict__ batch,
                                                   float* __restrict__ hout,
                                                   float* __restrict__ pool) {
    int gid  = blockIdx.x * blockDim.x + threadIdx.x;
    int node = gid >> 5;
    int c4   = (gid & 31) * 4;
    if (node >= N_NODES) return;

    float4 v  = *reinterpret_cast<const float4*>(h2 + (size_t)node * DIM + c4);
    float4 sc = *reinterpret_cast<const float4*>(ss + c4);
    float4 sh = *reinterpret_cast<const float4*>(ss + DIM + c4);
    v.x = v.x * sc.x + sh.x;
    v.y = v.y * sc.y + sh.y;
    v.z = v.z * sc.z + sh.z;
    v.w = v.w * sc.w + sh.w;
    *reinterpret_cast<float4*>(hout + (size_t)node * DIM + c4) = v;

    const int g = batch[node];
    float* p = pool + (size_t)g * (N_LAYERS * DIM) + c4;
    atomicAdd(p + 0, v.x);
    atomicAdd(p + 1, v.y);
    atomicAdd(p + 2, v.z);
    atomicAdd(p + 3, v.w);
}

// ---------------------------------------------------------------------------
extern "C" void kernel_launch(void* const* d_in, const int* in_sizes, int n_in,
                              void* d_out, int out_size, void* d_ws, size_t ws_size,
                              hipStream_t stream) {
    (void)in_sizes; (void)n_in; (void)ws_size;

    const float* x      = (const float*)d_in[0];
    const int*   ei     = (const int*)d_in[1];           // [2, N_EDGES]
    const int*   batch  = (const int*)d_in[2];
    const float* W1s    = (const float*)d_in[3];         // [3,128,128]
    const float* b1s    = (const float*)d_in[4];         // [3,128]
    const float* W2s    = (const float*)d_in[5];
    const float* b2s    = (const float*)d_in[6];
    const float* gammas = (const float*)d_in[7];
    const float* betas  = (const float*)d_in[8];
    float*       out    = (float*)d_out;                 // [512, 384]

    const int* src = ei;
    const int* dst = ei + N_EDGES;

    const size_t feat = (size_t)N_NODES * DIM;           // 6.4M floats
    float* agg   = (float*)d_ws;
    float* h1    = agg  + feat;
    float* h2    = h1   + feat;
    float* hbuf  = h2   + feat;
    float* stats = hbuf + feat;                          // sum[128], sumsq[128]
    float* ss    = stats + 2 * DIM;                      // scale[128], shift[128]

    const int feat4        = (int)(feat / 4);
    const int copy_blocks  = (feat4 + 255) / 256;                 // 6250
    const int scat_blocks  = (N_EDGES * 32 + 255) / 256;          // 75000
    const int gemm_blocks  = N_NODES / 16;                        // 3125 (exact)
    const int pool_blocks  = (N_NODES * 32 + 255) / 256;          // 6250

    // d_out is atomically accumulated -> zero it every call
    gin_zero<<<(out_size + 255) / 256, 256, 0, stream>>>(out, out_size);

    const float* h = x;                                  // layer-0 input
    for (int layer = 0; layer < N_LAYERS; ++layer) {
        const float* W1 = W1s + (size_t)layer * DIM * DIM;
        const float* b1 = b1s + (size_t)layer * DIM;
        const float* W2 = W2s + (size_t)layer * DIM * DIM;
        const float* b2 = b2s + (size_t)layer * DIM;
        const float* ga = gammas + (size_t)layer * DIM;
        const float* be = betas  + (size_t)layer * DIM;

        gin_zero<<<1, 256, 0, stream>>>(stats, 2 * DIM);

        // agg = h ; agg[dst] += h[src]
        gin_copy<<<copy_blocks, 256, 0, stream>>>(h, agg, feat4);
        gin_scatter<<<scat_blocks, 256, 0, stream>>>(h, src, dst, agg, N_EDGES);

        // h1 = relu(agg @ W1 + b1)
        gin_gemm_relu<<<gemm_blocks, 256, 0, stream>>>(agg, W1, b1, h1,
                                                       nullptr, nullptr);
        // h2 = relu(h1 @ W2 + b2) fused with BN column statistics
        gin_gemm_relu<<<gemm_blocks, 256, 0, stream>>>(h1, W2, b2, h2,
                                                       stats, stats + DIM);

        gin_bnparams<<<1, 128, 0, stream>>>(stats, ga, be, ss);

        // h = BN(h2); out[:, layer*128 : (layer+1)*128] += segment_sum(h, batch)
        gin_bn_pool<<<pool_blocks, 256, 0, stream>>>(h2, ss, batch, hbuf,
                                                     out + (size_t)layer * DIM);
        h = hbuf;
    }
}